// DynamicGWD_67113158967399
// MI455X (gfx1250) — compile-verified
//
#include <hip/hip_runtime.h>
#include <math.h>

#define TAU_F 0.3f
#define EPS_F 1e-8f

constexpr int THREADS          = 256;               // 8 waves (wave32)
constexpr int TILE             = 512;               // boxes per block-tile
constexpr int FLOATS_PER_TENS  = TILE * 7;          // 3584 floats / tensor / tile
constexpr int VEC4_PER_TENS    = FLOATS_PER_TENS/4; // 896
constexpr int VEC4_PER_TILE    = 2 * VEC4_PER_TENS; // 1792 -> exactly 7 per thread
constexpr int BUF_FLOATS       = 2 * FLOATS_PER_TENS; // 7168 floats = 28 KB / buffer

// Generic pointers into LDS carry the LDS byte offset in their low 32 bits
// (ISA 10.2 aperture mapping).
__device__ __forceinline__ unsigned lds_addr_of(const void* p) {
  return (unsigned)(unsigned long long)p;
}

// CDNA5 async global->LDS copy (ASYNCcnt-tracked, bypasses VGPR staging).
__device__ __forceinline__ void async_load_b128(unsigned lds_addr, const void* gaddr) {
  asm volatile("global_load_async_to_lds_b128 %0, %1, off"
               :: "v"(lds_addr), "v"(gaddr) : "memory");
}
// Partial / full drains of the async counter (immediates only).
#define WAIT_ASYNCCNT(n) asm volatile("s_wait_asynccnt " #n ::: "memory")

// Issue exactly 7 async b128 transfers per thread (uniform per wave!) copying
// one tile (pred block then gt block) into the LDS buffer at lds_base.
__device__ __forceinline__ void prefetch_tile(const float* __restrict__ pred,
                                              const float* __restrict__ gt,
                                              long long tile, unsigned lds_base,
                                              int tid, long long total4) {
  const long long f4base = tile * (long long)VEC4_PER_TENS;
  #pragma unroll
  for (int u = 0; u < VEC4_PER_TILE / THREADS; ++u) {   // 7 iterations
    const int i = tid + u * THREADS;
    const bool isGt = i >= VEC4_PER_TENS;
    const int  j    = isGt ? (i - VEC4_PER_TENS) : i;
    long long g4 = f4base + j;
    if (g4 >= total4) g4 = total4 - 1;                  // clamp: ALWAYS issue
    const float* src = (isGt ? gt : pred) + 4ll * g4;
    async_load_b128(lds_base + 16u * (unsigned)i, src);
  }
}

// Closed-form GWD loss for one box pair (values already in registers).
__device__ __forceinline__ void gwd_accum(
    float px, float py, float pz, float pdx, float pdy, float pdz, float pth,
    float tx, float ty, float tz, float tdx, float tdy, float tdz, float tth,
    float& accL, float& accC) {
  const float dx = px - tx, dy = py - ty, dz = pz - tz;
  const float mu2 = dx * dx + dy * dy + dz * dz;

  // sigma eigenvalues (rotation about z): {dx^2/4, dy^2/4, dz^2/4}
  const float ap = 0.25f * pdx * pdx, bpv = 0.25f * pdy * pdy, cp = 0.25f * pdz * pdz;
  const float at = 0.25f * tdx * tdx, btv = 0.25f * tdy * tdy, ct = 0.25f * tdz * tdz;

  // sqrtm_psd clamps eigenvalues at EPS before sqrt
  const float apc = fmaxf(ap, EPS_F), bpc = fmaxf(bpv, EPS_F), cpc = fmaxf(cp, EPS_F);

  float sn_p, cs_p, sn_t, cs_t;
  __sincosf(pth, &sn_p, &cs_p);
  __sincosf(tth, &sn_t, &cs_t);
  const float pc2 = cs_p * cs_p, ps2 = sn_p * sn_p, pcs = cs_p * sn_p;
  const float tc2 = cs_t * cs_t, ts2 = sn_t * sn_t, tcs = cs_t * sn_t;

  // 2x2 xy blocks: P = (sqrtm sigma_p)^2 (clamped), T = sigma_t
  const float P00 = apc * pc2 + bpc * ps2;
  const float P11 = apc * ps2 + bpc * pc2;
  const float P01 = (apc - bpc) * pcs;
  const float T00 = at * tc2 + btv * ts2;
  const float T11 = at * ts2 + btv * tc2;
  const float T01 = (at - btv) * tcs;

  // tr(sqrt(M)) for 2x2 PSD M = P^{1/2} T P^{1/2}:
  //   trM = <P,T>, detM = detP*detT, tr(sqrtM) = sqrt(trM + 2*sqrt(detM))
  const float trM  = P00 * T00 + 2.0f * P01 * T01 + P11 * T11;
  const float detM = (apc * bpc) * (at * btv);
  const float tr2  = sqrtf(fmaxf(trM + 2.0f * sqrtf(fmaxf(detM, 0.0f)), 0.0f));
  const float trz  = sqrtf(fmaxf(cpc * ct, EPS_F));   // zz block of 2nd sqrtm

  const float cov = (ap + bpv + cp) + (at + btv + ct) - 2.0f * (tr2 + trz);
  const float w2  = mu2 + cov;

  const float dett = fmaxf((at * btv) * ct, EPS_F);   // det(sigma_t)
  const float det3 = cbrtf(dett);

  const float loss = 1.0f - expf(-w2 / (TAU_F * det3));
  const float m = (tx != 0.0f) ? 1.0f : 0.0f;
  accL = fmaf(loss, m, accL);
  accC += m;
}

__global__ __launch_bounds__(THREADS)
void gwd_tile_kernel(const float* __restrict__ pred, const float* __restrict__ gt,
                     float* __restrict__ partials,
                     long long totalBoxes, long long totalFloats, int numTiles) {
  __shared__ alignas(16) float s_buf[2][BUF_FLOATS];   // double buffer, 56 KB
  __shared__ float s_red[2][THREADS];

  const int tid = threadIdx.x;
  const long long total4 = totalFloats >> 2;           // complete vec4s available
  const unsigned buf_base[2] = { lds_addr_of(s_buf[0]), lds_addr_of(s_buf[1]) };

  float accL = 0.0f, accC = 0.0f;

  long long tile = blockIdx.x;
  if (tile < numTiles) {
    // Prologue: start DMA for the first tile (batch of 7 per wave).
    prefetch_tile(pred, gt, tile, buf_base[0], tid, total4);

    int k = 0;
    for (; tile < numTiles; tile += gridDim.x, ++k) {
      const long long nxt = tile + gridDim.x;
      if (nxt < numTiles) {
        // Overlap: kick off next tile's DMA, then drain only the PREVIOUS
        // batch (in-order completion: cnt<=7 => first 7 have landed).
        prefetch_tile(pred, gt, nxt, buf_base[(k + 1) & 1], tid, total4);
        WAIT_ASYNCCNT(7);
      } else {
        WAIT_ASYNCCNT(0);
      }
      __syncthreads();                 // all waves' transfers visible in LDS

      const float* buf = s_buf[k & 1];
      #pragma unroll
      for (int u = 0; u < TILE / THREADS; ++u) {        // 2 boxes per thread
        const int local = tid + u * THREADS;
        const long long box = tile * (long long)TILE + local;
        if (box < totalBoxes) {
          float p0,p1,p2,p3,p4,p5,p6, g0,g1,g2,g3,g4,g5,g6;
          if (7ll * (box + 1) <= 4ll * total4) {
            // Fast path: box fully resident in LDS.
            const float* bp = &buf[7 * local];                    // stride 7: conflict-free
            const float* bg = &buf[FLOATS_PER_TENS + 7 * local];  // +3584 (== 0 mod 64 banks)
            p0=bp[0]; p1=bp[1]; p2=bp[2]; p3=bp[3]; p4=bp[4]; p5=bp[5]; p6=bp[6];
            g0=bg[0]; g1=bg[1]; g2=bg[2]; g3=bg[3]; g4=bg[4]; g5=bg[5]; g6=bg[6];
          } else {
            // Ragged end (totalFloats % 4 != 0): read straight from HBM.
            const float* bp = pred + 7ll * box;
            const float* bg = gt   + 7ll * box;
            p0=bp[0]; p1=bp[1]; p2=bp[2]; p3=bp[3]; p4=bp[4]; p5=bp[5]; p6=bp[6];
            g0=bg[0]; g1=bg[1]; g2=bg[2]; g3=bg[3]; g4=bg[4]; g5=bg[5]; g6=bg[6];
          }
          gwd_accum(p0,p1,p2,p3,p4,p5,p6, g0,g1,g2,g3,g4,g5,g6, accL, accC);
        }
      }
      __syncthreads();   // all reads done before this buffer is overwritten
    }
  }

  // Deterministic in-block tree reduction.
  s_red[0][tid] = accL;
  s_red[1][tid] = accC;
  __syncthreads();
  for (int s = THREADS >> 1; s > 0; s >>= 1) {
    if (tid < s) {
      s_red[0][tid] += s_red[0][tid + s];
      s_red[1][tid] += s_red[1][tid + s];
    }
    __syncthreads();
  }
  if (tid == 0) {
    partials[2 * blockIdx.x]     = s_red[0][0];
    partials[2 * blockIdx.x + 1] = s_red[1][0];
  }
}

__global__ __launch_bounds__(THREADS)
void gwd_final_kernel(const float* __restrict__ partials, int nBlocks,
                      float* __restrict__ out) {
  __shared__ float red[2][THREADS];
  float l = 0.0f, c = 0.0f;
  for (int i = threadIdx.x; i < nBlocks; i += THREADS) {
    l += partials[2 * i];
    c += partials[2 * i + 1];
  }
  red[0][threadIdx.x] = l;
  red[1][threadIdx.x] = c;
  __syncthreads();
  for (int s = THREADS >> 1; s > 0; s >>= 1) {
    if (threadIdx.x < s) {
      red[0][threadIdx.x] += red[0][threadIdx.x + s];
      red[1][threadIdx.x] += red[1][threadIdx.x + s];
    }
    __syncthreads();
  }
  if (threadIdx.x == 0) out[0] = red[0][0] / (red[1][0] + EPS_F);
}

extern "C" void kernel_launch(void* const* d_in, const int* in_sizes, int n_in,
                              void* d_out, int out_size, void* d_ws, size_t ws_size,
                              hipStream_t stream) {
  const float* pred = (const float*)d_in[0];
  const float* gt   = (const float*)d_in[1];
  float* out        = (float*)d_out;
  float* partials   = (float*)d_ws;

  const long long totalFloats = (long long)in_sizes[0];   // B*N*7
  const long long totalBoxes  = totalFloats / 7;
  int numTiles = (int)((totalBoxes + TILE - 1) / TILE);   // 2048 for B=16,N=65536

  int blocks = numTiles < 1024 ? numTiles : 1024;         // >=2 tiles/block -> pipeline active
  const long long maxWsBlocks = (long long)(ws_size / (2 * sizeof(float)));
  if (maxWsBlocks > 0 && blocks > maxWsBlocks) blocks = (int)maxWsBlocks;
  if (blocks < 1) blocks = 1;

  gwd_tile_kernel<<<blocks, THREADS, 0, stream>>>(pred, gt, partials,
                                                  totalBoxes, totalFloats, numTiles);
  gwd_final_kernel<<<1, THREADS, 0, stream>>>(partials, blocks, out);
}